// INFORMForTrainer_22909355557427
// MI455X (gfx1250) — compile-verified
//
#include <hip/hip_runtime.h>

// bias = (1/NNZ) * sum_e vals[e] * dot(x[rows[e]], x[cols[e]])
//
// Derivation: sum(x * Lx) = trace(x^T L x) = sum_e vals[e] * <x[rows[e]], x[cols[e]]>
// which removes the segment_sum scatter entirely. Memory-bound gather problem:
// x (51.2 MB) is L2-resident on MI455X (192 MB L2); we batch 16 edges per wave
// and run the dots through V_WMMA_F32_16X16X4_F32 (D = A * B^T over K=128 in 32
// steps), reading the 16 edge-dots off the diagonal of the 16x16 accumulator.

typedef float v2f __attribute__((ext_vector_type(2)));
typedef float v8f __attribute__((ext_vector_type(8)));

#define N_FEAT 128

__global__ __launch_bounds__(256)
void spmm_trace_wmma_kernel(const float* __restrict__ x,
                            const int*   __restrict__ rows,
                            const int*   __restrict__ cols,
                            const float* __restrict__ vals,
                            float*       __restrict__ partials,
                            int nnz) {
  const int lane        = threadIdx.x & 31;
  const int waveInBlock = threadIdx.x >> 5;
  const int wavesPerBlk = blockDim.x >> 5;

  // Force wave-uniform loop control into SGPRs (keeps EXEC all-ones around WMMA).
  const int globalWave = __builtin_amdgcn_readfirstlane(blockIdx.x * wavesPerBlk + waveInBlock);
  const int nWaves     = __builtin_amdgcn_readfirstlane(gridDim.x * wavesPerBlk);

  // WMMA f32 16x16x4 A-fragment layout (ISA 7.12.2):
  //   lanes 0-15 hold M=0..15; VGPR0 = K {0 | 2}, VGPR1 = K {1 | 3} (low|high half-wave).
  // B (4x16) mirrors this with N in lanes. Both gathers collapse to:
  //   lane L reads float2 at x[idx[L&15]][4*kc + 2*(L>>4)]
  const int m    = lane & 15;   // edge-in-tile (A: row side via rows[], B: col side via cols[])
  const int half = lane >> 4;   // selects K {0,1} vs K {2,3} of each K4 chunk

  const int nTiles = (nnz + 15) >> 4;

  float acc = 0.0f;

  for (int tile = globalWave; tile < nTiles; tile += nWaves) {
    const int e        = tile * 16 + m;
    const int eClamped = (e < nnz) ? e : 0;          // tail edges read edge 0, masked via vals
    const int r = rows[eClamped];
    const int c = cols[eClamped];
    const float* __restrict__ xr = x + (size_t)r * N_FEAT + half * 2;
    const float* __restrict__ xc = x + (size_t)c * N_FEAT + half * 2;

    v8f acc16 = {};                                   // 16x16 f32 C/D accumulator
#pragma unroll
    for (int kc = 0; kc < 32; ++kc) {
      v2f a = *(const v2f*)(xr + kc * 4);             // aligned 8B gathers, L2-resident
      v2f b = *(const v2f*)(xc + kc * 4);
      acc16 = __builtin_amdgcn_wmma_f32_16x16x4_f32(
          /*neg_a=*/false, a, /*neg_b=*/false, b,
          /*c_mod=*/(short)0, acc16, /*reuse_a=*/false, /*reuse_b=*/false);
    }

    // Diagonal (m,m) of C: VGPR v holds it in lane v (m=v) and lane v+24 (m=v+8).
    float diag = 0.0f;
#pragma unroll
    for (int v = 0; v < 8; ++v) {
      const bool sel = (lane == v) || (lane == v + 24);
      diag = sel ? acc16[v] : diag;
    }
    const bool holder = (lane < 8) || (lane >= 24);
    const int  mh     = (lane < 8) ? lane : (lane - 16);
    const int  eh     = tile * 16 + mh;
    const float w     = (holder && eh < nnz) ? vals[eh] : 0.0f;
    acc += diag * w;

    // Prefetch next tile's index data (global_prefetch_b8 path).
    const int tnext = tile + nWaves;
    if (tnext < nTiles) {
      __builtin_prefetch(&rows[tnext * 16 + m], 0, 0);
      __builtin_prefetch(&cols[tnext * 16 + m], 0, 0);
    }
  }

  // Wave32 butterfly reduction.
#pragma unroll
  for (int off = 16; off > 0; off >>= 1)
    acc += __shfl_xor(acc, off, 32);

  __shared__ float wsum[8];
  if (lane == 0) wsum[waveInBlock] = acc;
  __syncthreads();
  if (threadIdx.x == 0) {
    float s = 0.0f;
    for (int w = 0; w < wavesPerBlk; ++w) s += wsum[w];
    partials[blockIdx.x] = s;
  }
}

__global__ __launch_bounds__(256)
void reduce_partials_kernel(const float* __restrict__ partials, int n,
                            float* __restrict__ out, float invNNZ) {
  __shared__ float sm[256];
  float s = 0.0f;
  for (int i = threadIdx.x; i < n; i += blockDim.x) s += partials[i];
  sm[threadIdx.x] = s;
  __syncthreads();
  for (int stride = 128; stride > 0; stride >>= 1) {
    if ((int)threadIdx.x < stride) sm[threadIdx.x] += sm[threadIdx.x + stride];
    __syncthreads();
  }
  if (threadIdx.x == 0) out[0] = sm[0] * invNNZ;
}

extern "C" void kernel_launch(void* const* d_in, const int* in_sizes, int n_in,
                              void* d_out, int out_size, void* d_ws, size_t ws_size,
                              hipStream_t stream) {
  const float* x    = (const float*)d_in[0];
  const int*   rows = (const int*)  d_in[1];
  const int*   cols = (const int*)  d_in[2];
  const float* vals = (const float*)d_in[3];
  const int    nnz  = in_sizes[1];

  float* partials = (float*)d_ws;

  int nblocks = 2048;                                 // 16384 waves for latency hiding
  const size_t maxb = ws_size / sizeof(float);
  if ((size_t)nblocks > maxb) nblocks = (int)maxb;
  if (nblocks < 1) nblocks = 1;

  spmm_trace_wmma_kernel<<<nblocks, 256, 0, stream>>>(x, rows, cols, vals, partials, nnz);
  reduce_partials_kernel<<<1, 256, 0, stream>>>(partials, nblocks, (float*)d_out,
                                                1.0f / (float)nnz);
}